// MoleculeEncoder_20942260536133
// MI455X (gfx1250) — compile-verified
//
#include <hip/hip_runtime.h>
#include <hip/hip_bf16.h>

typedef float v2f __attribute__((ext_vector_type(2)));
typedef float v8f __attribute__((ext_vector_type(8)));

static __device__ __forceinline__ v8f wmma_f32_4(v2f a, v2f b, v8f c) {
    return __builtin_amdgcn_wmma_f32_16x16x4_f32(
               false, a, false, b, (short)0, c, false, false);
}

// ---------------------------------------------------------------------------
// Templated fp32 WMMA GEMM:  C[M,NC] = A[M,K] @ B[K,NC] + bias
// K, NC compile-time -> all inner-loop addresses are base + const-imm offsets.
// Block = 128 threads = 4 waves; wave w computes a 32x32 tile at
// (blockIdx.x*32, (blockIdx.y*4+w)*32): 4 accumulators share 2 A frags and
// 2 B frags -> 4 WMMAs per 32 loaded bytes per lane per k-step.
// Optionally accumulates per-column sum / sum-of-squares (BatchNorm stats).
// ---------------------------------------------------------------------------
template <int K, int NC>
__global__ __launch_bounds__(128)
void wmma_gemm_f32_t(const float* __restrict__ A,
                     const float* __restrict__ B,
                     const float* __restrict__ bias,
                     float* __restrict__ C,
                     int M,
                     float* __restrict__ colsum,
                     float* __restrict__ colsumsq)
{
    const int lane = threadIdx.x & 31;
    const int wave = threadIdx.x >> 5;
    const int half = lane >> 4;          // 0: lanes 0-15, 1: lanes 16-31
    const int l15  = lane & 15;
    const int row0 = blockIdx.x * 32;
    const int col0 = (blockIdx.y * 4 + wave) * 32;

    // Clamp A rows instead of predicating loads: EXEC stays all-1 in the loop.
    const int mrow0 = min(row0 + l15,      M - 1);
    const int mrow1 = min(row0 + 16 + l15, M - 1);
    const float* __restrict__ Arow0 = A + (size_t)mrow0 * K + 2 * half;
    const float* __restrict__ Arow1 = A + (size_t)mrow1 * K + 2 * half;
    // B fragment bases for the two 16-col subtiles (VGPR v -> K = k0+2*half+v)
    const float* __restrict__ Bp0 = B + (size_t)(2 * half) * NC + col0 + l15;
    const float* __restrict__ Bp1 = Bp0 + 16;

    v8f acc00 = {};   // rows [row0,row0+16) x cols [col0,col0+16)
    v8f acc01 = {};   // rows [row0,row0+16) x cols [col0+16,col0+32)
    v8f acc10 = {};   // rows [row0+16,row0+32) x cols [col0,col0+16)
    v8f acc11 = {};   // rows [row0+16,row0+32) x cols [col0+16,col0+32)
    #pragma unroll
    for (int k0 = 0; k0 < K; k0 += 4) {
        v2f a0 = *(const v2f*)(Arow0 + k0);     // 8B aligned (even index)
        v2f a1 = *(const v2f*)(Arow1 + k0);
        v2f b0, b1;
        b0.x = Bp0[(size_t)(k0    ) * NC];      // const-imm offsets
        b0.y = Bp0[(size_t)(k0 + 1) * NC];
        b1.x = Bp1[(size_t)(k0    ) * NC];
        b1.y = Bp1[(size_t)(k0 + 1) * NC];
        acc00 = wmma_f32_4(a0, b0, acc00);
        acc01 = wmma_f32_4(a0, b1, acc01);
        acc10 = wmma_f32_4(a1, b0, acc10);
        acc11 = wmma_f32_4(a1, b1, acc11);
    }

    const float bv0 = bias ? bias[col0 + l15]      : 0.0f;
    const float bv1 = bias ? bias[col0 + 16 + l15] : 0.0f;

    // C/D layout: VGPR r -> row rowbase + r + 8*half, col = colbase + l15
    float s0 = 0.0f, q0 = 0.0f, s1 = 0.0f, q1 = 0.0f;
    #pragma unroll
    for (int r = 0; r < 8; ++r) {
        const int rowA = row0 + r + 8 * half;        // row tile 0
        if (rowA < M) {
            const float v0 = acc00[r] + bv0;
            const float v1 = acc01[r] + bv1;
            float* crow = C + (size_t)rowA * NC + col0 + l15;
            crow[0]  = v0;
            crow[16] = v1;
            s0 += v0; q0 += v0 * v0;
            s1 += v1; q1 += v1 * v1;
        }
        const int rowB = rowA + 16;                  // row tile 1
        if (rowB < M) {
            const float v0 = acc10[r] + bv0;
            const float v1 = acc11[r] + bv1;
            float* crow = C + (size_t)rowB * NC + col0 + l15;
            crow[0]  = v0;
            crow[16] = v1;
            s0 += v0; q0 += v0 * v0;
            s1 += v1; q1 += v1 * v1;
        }
    }
    if (colsum) {   // fused BatchNorm statistics (4 atomics / 32 outputs per lane)
        atomicAdd(&colsum[col0 + l15],        s0);
        atomicAdd(&colsumsq[col0 + l15],      q0);
        atomicAdd(&colsum[col0 + 16 + l15],   s1);
        atomicAdd(&colsumsq[col0 + 16 + l15], q1);
    }
}

// ---------------------------------------------------------------------------
// Generic runtime-dim fallback (same math, 8 waves x one 16x16 tile).
// ---------------------------------------------------------------------------
__global__ __launch_bounds__(256)
void wmma_gemm_f32_g(const float* __restrict__ A,
                     const float* __restrict__ B,
                     const float* __restrict__ bias,
                     float* __restrict__ C,
                     int M, int Ncols, int K,
                     float* __restrict__ colsum,
                     float* __restrict__ colsumsq)
{
    const int lane = threadIdx.x & 31;
    const int wave = threadIdx.x >> 5;
    const int half = lane >> 4;
    const int l15  = lane & 15;
    const int row0 = blockIdx.x * 16;
    const int col0 = (blockIdx.y * 8 + wave) * 16;
    if (col0 >= Ncols) return;

    const int mrow = min(row0 + l15, M - 1);
    const float* Arow = A + (size_t)mrow * K + 2 * half;
    const float* Bp   = B + (size_t)(2 * half) * Ncols + col0 + l15;

    v8f acc = {};
    for (int k0 = 0; k0 < K; k0 += 4) {
        v2f a = *(const v2f*)(Arow + k0);
        v2f b;
        b.x = Bp[(size_t)(k0    ) * Ncols];
        b.y = Bp[(size_t)(k0 + 1) * Ncols];
        acc = wmma_f32_4(a, b, acc);
    }
    const float bv = bias ? bias[col0 + l15] : 0.0f;
    float s = 0.0f, sq = 0.0f;
    #pragma unroll
    for (int r = 0; r < 8; ++r) {
        const int row = row0 + r + 8 * half;
        if (row < M) {
            const float v = acc[r] + bv;
            C[(size_t)row * Ncols + col0 + l15] = v;
            s += v; sq += v * v;
        }
    }
    if (colsum) {
        atomicAdd(&colsum[col0 + l15], s);
        atomicAdd(&colsumsq[col0 + l15], sq);
    }
}

// z = (1 + eps[l]) * h        (float4 vectorized; n4 = N*H/4)
__global__ void scale_h_kernel(const float* __restrict__ h,
                               float* __restrict__ z,
                               const float* __restrict__ eps, int l, size_t n4)
{
    size_t i = (size_t)blockIdx.x * blockDim.x + threadIdx.x;
    if (i >= n4) return;
    const float e = 1.0f + eps[l];
    float4 v = ((const float4*)h)[i];
    v.x *= e; v.y *= e; v.z *= e; v.w *= e;
    ((float4*)z)[i] = v;
}

// z[row[e], :] += h[col[e], :]   (one thread per (edge, 4 features))
__global__ void scatter_add_kernel(const float* __restrict__ h,
                                   float* __restrict__ z,
                                   const int* __restrict__ rows,
                                   const int* __restrict__ cols,
                                   int E, int H)
{
    const int chunks = H >> 2;
    size_t i = (size_t)blockIdx.x * blockDim.x + threadIdx.x;
    if (i >= (size_t)E * chunks) return;
    const int e = (int)(i / chunks);
    const int c = (int)(i % chunks) << 2;
    const int src = cols[e];
    const int dst = rows[e];
    float4 v = *(const float4*)(h + (size_t)src * H + c);
    float* p = z + (size_t)dst * H + c;
    atomicAdd(p + 0, v.x);
    atomicAdd(p + 1, v.y);
    atomicAdd(p + 2, v.z);
    atomicAdd(p + 3, v.w);
}

// Fold BN stats into per-column scale/shift: y_norm = y*scale + shift
__global__ void bn_finalize_kernel(const float* __restrict__ colsum,
                                   const float* __restrict__ colsumsq,
                                   const float* __restrict__ gamma,
                                   const float* __restrict__ beta,
                                   float* __restrict__ scale,
                                   float* __restrict__ shift,
                                   int H, float invM)
{
    int c = blockIdx.x * blockDim.x + threadIdx.x;
    if (c >= H) return;
    const float mean = colsum[c] * invM;
    const float var  = colsumsq[c] * invM - mean * mean;
    const float sc   = gamma[c] * rsqrtf(var + 1e-5f);
    scale[c] = sc;
    shift[c] = beta[c] - mean * sc;
}

// dst = relu(y * scale[col] + shift[col])
__global__ void bn_relu_kernel(const float* __restrict__ y,
                               float* __restrict__ dst,
                               const float* __restrict__ scale,
                               const float* __restrict__ shift,
                               int H, size_t total)
{
    size_t i = (size_t)blockIdx.x * blockDim.x + threadIdx.x;
    if (i >= total) return;
    const int c = (int)(i % H);
    const float v = y[i] * scale[c] + shift[c];
    dst[i] = v > 0.0f ? v : 0.0f;
}

// pooled[batch[n], :] += h[n, :]
__global__ void pool_kernel(const float* __restrict__ h,
                            float* __restrict__ pooled,
                            const int* __restrict__ batch,
                            int N, int H)
{
    const int chunks = H >> 2;
    size_t i = (size_t)blockIdx.x * blockDim.x + threadIdx.x;
    if (i >= (size_t)N * chunks) return;
    const int n = (int)(i / chunks);
    const int c = (int)(i % chunks) << 2;
    const int g = batch[n];
    float4 v = *(const float4*)(h + (size_t)n * H + c);
    float* p = pooled + (size_t)g * H + c;
    atomicAdd(p + 0, v.x);
    atomicAdd(p + 1, v.y);
    atomicAdd(p + 2, v.z);
    atomicAdd(p + 3, v.w);
}

extern "C" void kernel_launch(void* const* d_in, const int* in_sizes, int n_in,
                              void* d_out, int out_size, void* d_ws, size_t ws_size,
                              hipStream_t stream)
{
    const float* x      = (const float*)d_in[0];
    const int*   edge   = (const int*)  d_in[1];   // [2, E]
    const int*   batch  = (const int*)  d_in[2];   // [N]
    const float* emb_W  = (const float*)d_in[4];
    const float* emb_b  = (const float*)d_in[5];
    const float* W1     = (const float*)d_in[6];
    const float* b1     = (const float*)d_in[7];
    const float* g1     = (const float*)d_in[8];
    const float* be1    = (const float*)d_in[9];
    const float* W2     = (const float*)d_in[10];
    const float* b2     = (const float*)d_in[11];
    const float* g2     = (const float*)d_in[12];
    const float* be2    = (const float*)d_in[13];
    const float* eps    = (const float*)d_in[14];
    const float* proj_W = (const float*)d_in[15];
    const float* proj_b = (const float*)d_in[16];

    const int N = in_sizes[2];             // 50000
    const int E = in_sizes[1] / 2;         // 600000
    const int A = in_sizes[0] / N;         // 64
    const int H = in_sizes[5];             // 128
    const int O = in_sizes[16];            // 256
    const int L = in_sizes[14];            // 3
    const int G = out_size / O;            // 1000

    // ---- workspace layout (floats) ----
    float* ws       = (float*)d_ws;
    float* h_buf    = ws;                                  // [N, H]
    float* y_buf    = h_buf + (size_t)N * H;               // [N, H]
    float* z_buf    = y_buf + (size_t)N * H;               // [N, H]
    float* pooled   = z_buf + (size_t)N * H;               // [G, H]
    float* colsum   = pooled + (size_t)G * H;              // [H]
    float* colsumsq = colsum + H;                          // [H]
    float* bn_scale = colsumsq + H;                        // [H]
    float* bn_shift = bn_scale + H;                        // [H]

    auto gemm = [&](const float* Ain, const float* Bin, const float* bias,
                    float* Cout, int M, int Ncols, int K,
                    float* cs, float* csq) {
        const int mt32 = (M + 31) / 32;
        if (K == 64 && Ncols == 128) {
            wmma_gemm_f32_t<64, 128><<<dim3(mt32, 1), 128, 0, stream>>>(
                Ain, Bin, bias, Cout, M, cs, csq);
        } else if (K == 128 && Ncols == 128) {
            wmma_gemm_f32_t<128, 128><<<dim3(mt32, 1), 128, 0, stream>>>(
                Ain, Bin, bias, Cout, M, cs, csq);
        } else if (K == 128 && Ncols == 256) {
            wmma_gemm_f32_t<128, 256><<<dim3(mt32, 2), 128, 0, stream>>>(
                Ain, Bin, bias, Cout, M, cs, csq);
        } else {
            wmma_gemm_f32_g<<<dim3((M + 15) / 16, (Ncols + 127) / 128),
                              256, 0, stream>>>(
                Ain, Bin, bias, Cout, M, Ncols, K, cs, csq);
        }
    };

    const dim3 blk(256);
    const size_t NH   = (size_t)N * H;
    const size_t n4   = NH / 4;
    const size_t echk = (size_t)E * (H / 4);
    const size_t pchk = (size_t)N * (H / 4);
    const float  invN = 1.0f / (float)N;

    // 1) h = x @ emb_W + emb_b
    gemm(x, emb_W, emb_b, h_buf, N, H, A, nullptr, nullptr);

    // 2) GIN layers
    for (int l = 0; l < L; ++l) {
        // z = (1+eps[l]) * h
        scale_h_kernel<<<(unsigned)((n4 + 255) / 256), blk, 0, stream>>>(
            h_buf, z_buf, eps, l, n4);
        // z[row] += h[col]  (segment_sum over edges)
        scatter_add_kernel<<<(unsigned)((echk + 255) / 256), blk, 0, stream>>>(
            h_buf, z_buf, edge, edge + E, E, H);

        // y = z @ W1[l] + b1[l]  (+ column stats)
        hipMemsetAsync(colsum, 0, 2 * H * sizeof(float), stream);
        gemm(z_buf, W1 + (size_t)l * H * H, b1 + (size_t)l * H,
             y_buf, N, H, H, colsum, colsumsq);
        bn_finalize_kernel<<<(H + 127) / 128, 128, 0, stream>>>(
            colsum, colsumsq, g1 + (size_t)l * H, be1 + (size_t)l * H,
            bn_scale, bn_shift, H, invN);
        bn_relu_kernel<<<(unsigned)((NH + 255) / 256), blk, 0, stream>>>(
            y_buf, z_buf, bn_scale, bn_shift, H, NH);

        // h = relu(bn(z @ W2[l] + b2[l]))
        hipMemsetAsync(colsum, 0, 2 * H * sizeof(float), stream);
        gemm(z_buf, W2 + (size_t)l * H * H, b2 + (size_t)l * H,
             y_buf, N, H, H, colsum, colsumsq);
        bn_finalize_kernel<<<(H + 127) / 128, 128, 0, stream>>>(
            colsum, colsumsq, g2 + (size_t)l * H, be2 + (size_t)l * H,
            bn_scale, bn_shift, H, invN);
        bn_relu_kernel<<<(unsigned)((NH + 255) / 256), blk, 0, stream>>>(
            y_buf, h_buf, bn_scale, bn_shift, H, NH);
    }

    // 3) pooled = segment_sum(h, batch_index)
    hipMemsetAsync(pooled, 0, (size_t)G * H * sizeof(float), stream);
    pool_kernel<<<(unsigned)((pchk + 255) / 256), blk, 0, stream>>>(
        h_buf, pooled, batch, N, H);

    // 4) out = pooled @ proj_W + proj_b
    gemm(pooled, proj_W, proj_b, (float*)d_out, G, O, H, nullptr, nullptr);
}